// GCN_with_glob_16673063043611
// MI455X (gfx1250) — compile-verified
//
#include <hip/hip_runtime.h>
#include <hip/hip_bf16.h>
#include <float.h>

typedef __attribute__((ext_vector_type(2))) float v2f;
typedef __attribute__((ext_vector_type(8))) float v8f;

#define NG 64   // num graphs
#define DD 64   // conv/global dim

// ---------------- small utility kernels ----------------

__global__ void k_fill(float* __restrict__ p, int n, float v) {
  int i = blockIdx.x * blockDim.x + threadIdx.x;
  if (i < n) p[i] = v;
}

__global__ void k_deg_hist(const int* __restrict__ col, float* __restrict__ deg, int E) {
  int e = blockIdx.x * blockDim.x + threadIdx.x;
  if (e < E) atomicAdd(&deg[col[e]], 1.0f);
}

__global__ void k_rsqrt_inplace(float* __restrict__ d, int n) {
  int i = blockIdx.x * blockDim.x + threadIdx.x;
  if (i < n) {
    float v = d[i];
    d[i] = (v > 0.f) ? rsqrtf(v) : 0.f;
  }
}

__global__ void k_norm(const int* __restrict__ row, const int* __restrict__ col,
                       const float* __restrict__ dinv, float* __restrict__ norm, int E) {
  int e = blockIdx.x * blockDim.x + threadIdx.x;
  if (e < E) norm[e] = dinv[row[e]] * dinv[col[e]];
}

// out[g,j] = gin[g(or 0),:] @ w + b  (+ parts[g,:] @ w_ng + b_ng  when parts != null)
__global__ void k_globs(const float* __restrict__ gin, int gin_stride,
                        const float* __restrict__ w, const float* __restrict__ b,
                        const float* __restrict__ parts,
                        const float* __restrict__ w_ng, const float* __restrict__ b_ng,
                        float* __restrict__ out) {
  int idx = blockIdx.x * blockDim.x + threadIdx.x;
  if (idx >= NG * DD) return;
  int g = idx >> 6, j = idx & 63;
  const float* grow = gin + g * gin_stride;
  float acc = b[j];
  for (int k = 0; k < DD; ++k) acc += grow[k] * w[k * DD + j];
  if (parts) {
    acc += b_ng[j];
    const float* prow = parts + g * DD;
    for (int k = 0; k < DD; ++k) acc += prow[k] * w_ng[k * DD + j];
  }
  out[idx] = acc;
}

// ---------------- WMMA GEMM: out[N,64] = (relu?)A[N,K] @ W[K,64] + bias + gtab[gidx[n]] ----------------

template <bool RELU>
__global__ __launch_bounds__(256)
void k_gemm_wmma(const float* __restrict__ A, int K,
                 const float* __restrict__ W,
                 const float* __restrict__ bias,
                 const float* __restrict__ gtab,
                 const int* __restrict__ gidx,
                 float* __restrict__ out, int nNodes) {
  __shared__ float wlds[128 * 64];
  const int tid = threadIdx.x;
  const int nw = K * 64;
  for (int i = tid * 4; i < nw; i += 256 * 4)
    *(float4*)(&wlds[i]) = *(const float4*)(&W[i]);
  __syncthreads();

  const int wave = tid >> 5;
  const int lane = tid & 31;
  const int row0 = (blockIdx.x * 8 + wave) * 16;
  if (row0 + 16 > nNodes) return;   // wave-uniform: EXEC stays all-ones for WMMA

  const int mrow = lane & 15;
  const int koff = (lane >> 4) << 1;               // 0 for lanes 0-15, 2 for lanes 16-31
  const float* __restrict__ arow = A + (size_t)(row0 + mrow) * K;

  v8f c0 = {}; v8f c1 = {}; v8f c2 = {}; v8f c3 = {};
  for (int kk = 0; kk < K; kk += 4) {
    // A frag per ISA 16x4 f32 layout
    v2f a;
    a.x = arow[kk + koff];
    a.y = arow[kk + koff + 1];
    if (RELU) { a.x = fmaxf(a.x, 0.f); a.y = fmaxf(a.y, 0.f); }
    // B frags per ISA 4x16 f32 layout (rows striped across lanes)
    const float* wp = &wlds[(kk + koff) * 64 + mrow];
    v2f b0, b1, b2, b3;
    b0.x = wp[0];  b0.y = wp[64];
    b1.x = wp[16]; b1.y = wp[64 + 16];
    b2.x = wp[32]; b2.y = wp[64 + 32];
    b3.x = wp[48]; b3.y = wp[64 + 48];
    c0 = __builtin_amdgcn_wmma_f32_16x16x4_f32(false, a, false, b0, (short)0, c0, false, false);
    c1 = __builtin_amdgcn_wmma_f32_16x16x4_f32(false, a, false, b1, (short)0, c1, false, false);
    c2 = __builtin_amdgcn_wmma_f32_16x16x4_f32(false, a, false, b2, (short)0, c2, false, false);
    c3 = __builtin_amdgcn_wmma_f32_16x16x4_f32(false, a, false, b3, (short)0, c3, false, false);
  }

  // D layout: VGPR v -> row M=v (lanes 0-15) / M=v+8 (lanes 16-31), N = lane&15 within tile
  const int half8 = (lane >> 4) * 8;
  const int ncol = lane & 15;
  for (int v = 0; v < 8; ++v) {
    int r = row0 + v + half8;
    int g = gidx[r];
    float* orow = out + (size_t)r * 64;
    const float* grow = gtab + g * 64;
    orow[ncol +  0] = c0[v] + bias[ncol +  0] + grow[ncol +  0];
    orow[ncol + 16] = c1[v] + bias[ncol + 16] + grow[ncol + 16];
    orow[ncol + 32] = c2[v] + bias[ncol + 32] + grow[ncol + 32];
    orow[ncol + 48] = c3[v] + bias[ncol + 48] + grow[ncol + 48];
  }
}

// ---------------- SpMM aggregation ----------------

// agg[n,:] = dinv[n]^2 * h[n,:]   (self-loop contribution; also initializes agg)
__global__ void k_agg_init(const float* __restrict__ h, const float* __restrict__ dinv,
                           float* __restrict__ agg, int nNodes) {
  int idx = blockIdx.x * blockDim.x + threadIdx.x;   // n*16 + q
  int n = idx >> 4;
  if (n >= nNodes) return;
  float d = dinv[n];
  float s = d * d;
  int q = (idx & 15) * 4;
  float4 hv = *(const float4*)(h + (size_t)n * 64 + q);
  float4 o = make_float4(s * hv.x, s * hv.y, s * hv.z, s * hv.w);
  *(float4*)(agg + (size_t)n * 64 + q) = o;
}

// edges: agg[col,:] += norm[e] * h[row,:]
__global__ void k_scatter(const int* __restrict__ row, const int* __restrict__ col,
                          const float* __restrict__ norm, const float* __restrict__ h,
                          float* __restrict__ agg, int E) {
  int idx = blockIdx.x * blockDim.x + threadIdx.x;   // e*16 + q
  int e = idx >> 4;
  if (e >= E) return;
  int q = (idx & 15) * 4;
  int r = row[e], c = col[e];
  float nm = norm[e];
  float4 hv = *(const float4*)(h + (size_t)r * 64 + q);
  float* dst = agg + (size_t)c * 64 + q;
  atomicAdd(dst + 0, nm * hv.x);
  atomicAdd(dst + 1, nm * hv.y);
  atomicAdd(dst + 2, nm * hv.z);
  atomicAdd(dst + 3, nm * hv.w);
}

// ---------------- segment max (graph_indices sorted) ----------------

__device__ __forceinline__ void atomicMaxFloat(float* addr, float val) {
  if (val >= 0.f) atomicMax((int*)addr, __float_as_int(val));
  else            atomicMin((unsigned int*)addr, __float_as_uint(val));
}

__global__ void k_segmax(const float* __restrict__ agg, const int* __restrict__ gidx,
                         float* __restrict__ parts, int nNodes) {
  // blockDim = 64 (one thread per feature); each block scans 128 contiguous nodes
  int f = threadIdx.x;
  int n0 = blockIdx.x * 128;
  if (n0 >= nNodes) return;
  int n1 = min(n0 + 128, nNodes);
  int gcur = gidx[n0];
  float m = -FLT_MAX;
  for (int n = n0; n < n1; ++n) {
    int g = gidx[n];
    if (g != gcur) {
      atomicMaxFloat(&parts[gcur * DD + f], m);
      gcur = g;
      m = -FLT_MAX;
    }
    m = fmaxf(m, agg[(size_t)n * 64 + f]);
  }
  atomicMaxFloat(&parts[gcur * DD + f], m);
}

__global__ void k_sigmoid(float* __restrict__ d, int n) {
  int i = blockIdx.x * blockDim.x + threadIdx.x;
  if (i < n) d[i] = 1.f / (1.f + expf(-d[i]));
}

// ---------------- host launcher ----------------

static inline size_t align256(size_t x) { return (x + 255) & ~(size_t)255; }

extern "C" void kernel_launch(void* const* d_in, const int* in_sizes, int n_in,
                              void* d_out, int out_size, void* d_ws, size_t ws_size,
                              hipStream_t stream) {
  const float* x        = (const float*)d_in[0];
  const int*   ei       = (const int*)d_in[1];
  const int*   gidx     = (const int*)d_in[2];
  const float* glob0    = (const float*)d_in[3];
  const float* w_nn1    = (const float*)d_in[4];  const float* b_nn1 = (const float*)d_in[5];
  const float* w_gn1    = (const float*)d_in[6];  const float* b_gn1 = (const float*)d_in[7];
  const float* w_gg1    = (const float*)d_in[8];  const float* b_gg1 = (const float*)d_in[9];
  const float* w_ng1    = (const float*)d_in[10]; const float* b_ng1 = (const float*)d_in[11];
  const float* w_nn2    = (const float*)d_in[12]; const float* b_nn2 = (const float*)d_in[13];
  const float* w_gn2    = (const float*)d_in[14]; const float* b_gn2 = (const float*)d_in[15];
  // w_gg2/b_gg2/w_ng2/b_ng2 (d_in[16..19]) are dead: layer-2 global output is not returned.

  const int N  = in_sizes[2];           // 100000
  const int E  = in_sizes[1] / 2;       // 1600000
  const int K1 = in_sizes[0] / N;       // 128
  const int* row = ei;
  const int* col = ei + E;

  float* ws = (float*)d_ws;
  size_t o = 0;
  float* dinv   = ws + o; o += align256((size_t)N);
  float* norm   = ws + o; o += align256((size_t)E);
  float* bufA   = ws + o; o += (size_t)N * 64;     // h1, then h2
  float* bufB   = ws + o; o += (size_t)N * 64;     // agg1 (pre-relu)
  float* globs1 = ws + o; o += NG * DD;
  float* globl1 = ws + o; o += NG * DD;
  float* globs2 = ws + o; o += NG * DD;
  float* parts  = ws + o; o += NG * DD;
  (void)ws_size; (void)n_in; (void)out_size;

  float* outp = (float*)d_out;

  const int waves  = (N + 15) / 16;
  const int gGemm  = (waves + 7) / 8;
  const int gNode  = (N + 255) / 256;
  const int gEdge  = (E + 255) / 256;
  const int gNode16 = (N * 16 + 255) / 256;
  const int gEdge16 = (E * 16 + 255) / 256;
  const int gNF     = (N * 64 + 255) / 256;

  // degree -> dinv -> per-edge norm (reused by both layers)
  k_fill<<<gNode, 256, 0, stream>>>(dinv, N, 1.0f);                    // self loop
  k_deg_hist<<<gEdge, 256, 0, stream>>>(col, dinv, E);
  k_rsqrt_inplace<<<gNode, 256, 0, stream>>>(dinv, N);
  k_norm<<<gEdge, 256, 0, stream>>>(row, col, dinv, norm, E);

  // ---- layer 1 ----
  k_globs<<<16, 256, 0, stream>>>(glob0, 0, w_gn1, b_gn1, nullptr, nullptr, nullptr, globs1);
  k_gemm_wmma<false><<<gGemm, 256, 0, stream>>>(x, K1, w_nn1, b_nn1, globs1, gidx, bufA, N);
  k_agg_init<<<gNode16, 256, 0, stream>>>(bufA, dinv, bufB, N);
  k_scatter<<<gEdge16, 256, 0, stream>>>(row, col, norm, bufA, bufB, E);
  k_fill<<<16, 256, 0, stream>>>(parts, NG * DD, -FLT_MAX);
  k_segmax<<<(N + 127) / 128, 64, 0, stream>>>(bufB, gidx, parts, N);
  k_globs<<<16, 256, 0, stream>>>(glob0, 0, w_gg1, b_gg1, parts, w_ng1, b_ng1, globl1);

  // ---- layer 2 ----
  k_globs<<<16, 256, 0, stream>>>(globl1, DD, w_gn2, b_gn2, nullptr, nullptr, nullptr, globs2);
  k_gemm_wmma<true><<<gGemm, 256, 0, stream>>>(bufB, 64, w_nn2, b_nn2, globs2, gidx, bufA, N);
  k_agg_init<<<gNode16, 256, 0, stream>>>(bufA, dinv, outp, N);
  k_scatter<<<gEdge16, 256, 0, stream>>>(row, col, norm, bufA, outp, E);
  k_sigmoid<<<gNF, 256, 0, stream>>>(outp, N * 64);
}